// Adder_87531433492717
// MI455X (gfx1250) — compile-verified
//
#include <hip/hip_runtime.h>
#include <stdint.h>

// Base-10 ripple-carry addition of 524288 pairs of 64-digit numbers.
// Memory-bound: 384 MB traffic -> ~16.5 us floor at 23.3 TB/s.
// Stage A: coalesced CDNA5 async DMA (global -> LDS, ASYNCcnt) of a/b tiles,
//          XOR-swizzled LDS layout (bank-conflict-minimal, zero padding waste).
// Stage B: per-thread ripple-carry scan from LDS, direct b128 global stores.

namespace {
constexpr int SEQ        = 64;
constexpr int TILE_ROWS  = 128;
constexpr int THREADS    = 128;                  // 4 waves; thread t owns row t
constexpr int GROUPS     = SEQ / 4;              // 16 float4 groups per row
constexpr int ROW_BYTES  = SEQ * 4;              // 256 B, natural stride
constexpr uint32_t TILE_BYTES = (uint32_t)TILE_ROWS * ROW_BYTES;   // 32768
constexpr int XFERS = (TILE_ROWS * GROUPS) / THREADS;              // 16 per array
}

// 2 tiles x 32 KB = exactly 64 KB static LDS.
extern "C" __global__ void __launch_bounds__(THREADS)
adder_carry_scan(const float* __restrict__ a,
                 const float* __restrict__ b,
                 float* __restrict__ out)
{
    __shared__ float lds[2 * TILE_ROWS * SEQ];

    const int t  = threadIdx.x;
    const int tx = t & 15;          // group slot owned by this thread in stage A
    const int r0 = t >> 4;          // starting tile row for this thread's DMAs
    const long long tileRow0 = (long long)blockIdx.x * TILE_ROWS;

    // ---- Stage A: async DMA both tiles into LDS (ASYNCcnt path).
    // Transfer i = t + k*THREADS moves one 16B group: row r = r0 + 8k, group jg = tx.
    // Global side: 32 consecutive lanes touch 512 consecutive bytes -> fully coalesced.
    // LDS side: group stored at swizzled slot (jg ^ (r & 15)) within its row.
    const uint32_t ldsBase = (uint32_t)(uintptr_t)lds;  // low 32 bits = LDS offset
    const char* gaBase = (const char*)a + tileRow0 * ROW_BYTES + t * 16;
    const char* gbBase = (const char*)b + tileRow0 * ROW_BYTES + t * 16;

    #pragma unroll
    for (int k = 0; k < XFERS; ++k) {
        const int r   = r0 + 8 * k;
        const int swz = (tx ^ (r & 15)) * 16;
        const uint32_t loff = (uint32_t)(r * ROW_BYTES + swz);
        uint32_t la = ldsBase + loff;
        uint32_t lb = ldsBase + TILE_BYTES + loff;
        unsigned long long ga = (unsigned long long)(uintptr_t)(gaBase + k * (THREADS * 16));
        unsigned long long gb = (unsigned long long)(uintptr_t)(gbBase + k * (THREADS * 16));

        asm volatile("global_load_async_to_lds_b128 %0, %1, off"
                     :: "v"(la), "v"(ga) : "memory");
        asm volatile("global_load_async_to_lds_b128 %0, %1, off"
                     :: "v"(lb), "v"(gb) : "memory");
    }
    asm volatile("s_wait_asynccnt 0" ::: "memory");
    __syncthreads();

    // ---- Stage B: ripple-carry scan, LSB (j=63) -> MSB (j=0).
    // Chunk c of row t lives at swizzled slot (c ^ tx). Per b128 wave read the
    // 32 lanes hit all 64 banks with 2 lanes per 4-bank group = 2-cycle minimum.
    const float* arow = lds + t * SEQ;
    const float* brow = lds + (TILE_BYTES / 4) + t * SEQ;
    float4* __restrict__ orow = (float4*)(out + (tileRow0 + t) * SEQ);

    float carry = 0.0f;
    #pragma unroll
    for (int c = GROUPS - 1; c >= 0; --c) {
        const int sc = (c ^ tx) * 4;
        const float4 a4 = *(const float4*)(arow + sc);
        const float4 b4 = *(const float4*)(brow + sc);
        float4 d4;
        float s;
        s = a4.w + b4.w + carry; carry = (s >= 10.0f) ? 1.0f : 0.0f; d4.w = s - 10.0f * carry;
        s = a4.z + b4.z + carry; carry = (s >= 10.0f) ? 1.0f : 0.0f; d4.z = s - 10.0f * carry;
        s = a4.y + b4.y + carry; carry = (s >= 10.0f) ? 1.0f : 0.0f; d4.y = s - 10.0f * carry;
        s = a4.x + b4.x + carry; carry = (s >= 10.0f) ? 1.0f : 0.0f; d4.x = s - 10.0f * carry;
        orow[c] = d4;   // block output = one contiguous 32 KB span -> L2 packs HBM lines
    }
}

extern "C" void kernel_launch(void* const* d_in, const int* in_sizes, int n_in,
                              void* d_out, int out_size, void* d_ws, size_t ws_size,
                              hipStream_t stream) {
    const float* a  = (const float*)d_in[0];
    const float* b  = (const float*)d_in[1];
    float* out      = (float*)d_out;

    const int batch  = in_sizes[0] / SEQ;        // 524288
    const int blocks = batch / TILE_ROWS;        // 4096 (exact)

    adder_carry_scan<<<blocks, THREADS, 0, stream>>>(a, b, out);
}